// GATLayer_regular_78812649882202
// MI455X (gfx1250) — compile-verified
//
#include <hip/hip_runtime.h>
#include <hip/hip_bf16.h>
#include <math.h>

typedef float v2f __attribute__((ext_vector_type(2)));
typedef float v8f __attribute__((ext_vector_type(8)));

#define D 128            // D_IN == D_OUT == 128

// ---------------------------------------------------------------------------
// Kernel A: initialize the attention-logit accumulators with their biases.
// ---------------------------------------------------------------------------
__global__ void gat_init_a(const float* __restrict__ ba1,
                           const float* __restrict__ ba2,
                           float* __restrict__ a1,
                           float* __restrict__ a2,
                           int n) {
    int i = blockIdx.x * blockDim.x + threadIdx.x;
    if (i < n) {
        a1[i] = ba1[0];
        a2[i] = ba2[0];
    }
}

// ---------------------------------------------------------------------------
// Kernel B: fused  out = relu(X @ W^T + b)  and  avec += (out @ wa)
// using V_WMMA_F32_16X16X4_F32 (fp32 in, fp32 acc).
//
// Block = 256 threads = 8 wave32s. Block handles a 16-row M-tile; wave w
// handles output columns [16w, 16w+16). K = 128 -> 32 WMMA steps of K=4.
//
// A-frag (16x4 f32, per ISA 7.12.2): lanes 0-15 hold M=lane, VGPR0=K0,
// VGPR1=K1; lanes 16-31 hold M=lane-16, VGPR0=K2, VGPR1=K3.
// B-frag (4x16): VGPR0 = row K0 (lanes 0-15, N=lane) / row K2 (lanes 16-31);
// VGPR1 = rows K1/K3.  B[k][n] = W[n*128 + k] (we need X @ W^T).
// C/D (16x16 f32): VGPR r -> M = r (lanes 0-15) / r+8 (lanes 16-31), N = lane%16.
// ---------------------------------------------------------------------------
__global__ __launch_bounds__(256)
void gat_linear_wmma(const float* __restrict__ feat,   // [N, 128]
                     const float* __restrict__ W,      // [128, 128] row-major (out, in)
                     const float* __restrict__ bias,   // [128]
                     const float* __restrict__ wa,     // [128]
                     float* __restrict__ outmat,       // [N, 128]
                     float* __restrict__ avec) {       // [N], pre-seeded with ba
    __shared__ float As[16 * D];   // 8 KB A-tile, shared by all 8 waves

    const int t  = threadIdx.x;
    const int m0 = blockIdx.x * 16;

    // Cooperative stage of the 16x128 fp32 A tile into LDS.
    {
        const float* src = feat + (size_t)m0 * D;
#pragma unroll
        for (int i = 0; i < 8; ++i)
            As[t + i * 256] = src[t + i * 256];
    }
    __syncthreads();

    const int wave = t >> 5;
    const int lane = t & 31;
    const int half = lane >> 4;    // 0: lanes 0-15, 1: lanes 16-31
    const int col  = lane & 15;    // N (and A-row M) index within tile
    const int n    = wave * 16 + col;

    const float* arow = As + col * D + 2 * half;            // A[M=col, K...]
    const float* brow = W + (size_t)n * D + 2 * half;       // B[k][n] = W[n][k]

    v8f c = {};
#pragma unroll
    for (int k0 = 0; k0 < D; k0 += 4) {
        v2f a = *(const v2f*)(arow + k0);
        v2f b = *(const v2f*)(brow + k0);
        // (neg_a, A, neg_b, B, c_mod, C, reuse_a, reuse_b)
        c = __builtin_amdgcn_wmma_f32_16x16x4_f32(false, a, false, b,
                                                  (short)0, c, false, false);
    }

    const float bn = bias[n];
    const float wn = wa[n];

#pragma unroll
    for (int r = 0; r < 8; ++r) {
        const int row = m0 + r + 8 * half;
        float v = fmaxf(c[r] + bn, 0.0f);          // bias + ReLU
        outmat[(size_t)row * D + n] = v;

        // partial dot with wa over this wave's 16 columns; reduce within
        // each 16-lane half (masks 1,2,4,8 never cross the half boundary).
        float p = v * wn;
        p += __shfl_xor(p, 1, 32);
        p += __shfl_xor(p, 2, 32);
        p += __shfl_xor(p, 4, 32);
        p += __shfl_xor(p, 8, 32);
        if (col == 0)
            atomicAdd(&avec[row], p);              // 8 waves x 16 cols = 128
    }
}

// ---------------------------------------------------------------------------
// Kernel C: edge scatter (incl. self loops). One wave32 per edge:
//   att = sigmoid(a1[r] + a2[c]);  out[r, :] += att * fj[c, :]
// Each lane owns 4 of the 128 columns (float4 gather + 4 f32 atomics).
// ---------------------------------------------------------------------------
__global__ __launch_bounds__(256)
void gat_edge_scatter(const int* __restrict__ rows,
                      const int* __restrict__ cols,
                      int E, int N,
                      const float* __restrict__ a1,
                      const float* __restrict__ a2,
                      const float* __restrict__ fj,
                      float* __restrict__ out) {
    const int warp = (blockIdx.x * blockDim.x + threadIdx.x) >> 5;
    const int lane = threadIdx.x & 31;
    const int total = E + N;
    if (warp >= total) return;

    int r, c;
    if (warp < E) { r = rows[warp]; c = cols[warp]; }
    else          { r = c = warp - E; }            // appended self-loop

    float z = a1[r] + a2[c];
    float att = 1.0f / (1.0f + __expf(-z));        // sigmoid

    const float4 v = ((const float4*)(fj + (size_t)c * D))[lane];
    float* dst = out + (size_t)r * D + lane * 4;
    atomicAdd(dst + 0, att * v.x);
    atomicAdd(dst + 1, att * v.y);
    atomicAdd(dst + 2, att * v.z);
    atomicAdd(dst + 3, att * v.w);
}

// ---------------------------------------------------------------------------
// Launch
// ---------------------------------------------------------------------------
extern "C" void kernel_launch(void* const* d_in, const int* in_sizes, int n_in,
                              void* d_out, int out_size, void* d_ws, size_t ws_size,
                              hipStream_t stream) {
    const float* feat = (const float*)d_in[0];   // [N,128]
    const float* W1   = (const float*)d_in[1];   // [128,128]
    const float* b1   = (const float*)d_in[2];   // [128]
    const float* W2   = (const float*)d_in[3];   // [128,128]
    const float* b2   = (const float*)d_in[4];   // [128]
    const float* wa1  = (const float*)d_in[5];   // [128]
    const float* ba1  = (const float*)d_in[6];   // [1]
    const float* wa2  = (const float*)d_in[7];   // [128]
    const float* ba2  = (const float*)d_in[8];   // [1]
    const int*   eidx = (const int*)d_in[9];     // [2,E] flattened

    const int N = in_sizes[0] / D;               // 50000
    const int E = in_sizes[9] / 2;               // 800000

    float* out = (float*)d_out;                  // [N,128] -> fi + agg
    float* fj  = (float*)d_ws;                   // [N,128]
    float* a1  = fj + (size_t)N * D;             // [N]
    float* a2  = a1 + N;                         // [N]

    // A: seed a1/a2 with scalar biases
    gat_init_a<<<(N + 255) / 256, 256, 0, stream>>>(ba1, ba2, a1, a2, N);

    // B: fi -> out (+a1), fj -> ws (+a2).  N is a multiple of 16 (50000/16=3125).
    const int mtiles = N / 16;
    gat_linear_wmma<<<mtiles, 256, 0, stream>>>(feat, W1, b1, wa1, out, a1);
    gat_linear_wmma<<<mtiles, 256, 0, stream>>>(feat, W2, b2, wa2, fj, a2);

    // C: scatter E real edges + N self loops, one wave per edge
    const int total = E + N;
    const int blocks = (total + 7) / 8;          // 8 waves per 256-thread block
    gat_edge_scatter<<<blocks, 256, 0, stream>>>(eidx, eidx + E, E, N,
                                                 a1, a2, fj, out);
}